// AtomicConvolution_29746943492208
// MI455X (gfx1250) — compile-verified
//
#include <hip/hip_runtime.h>
#include <hip/hip_bf16.h>
#include <stdint.h>

// Problem constants (match reference setup_inputs).
#define BB  16      // batch
#define NN  2048    // atoms per batch
#define MM  64      // neighbors per atom
#define LL  12      // radial symmetry params
#define CH  48      // 4 atom types * 12 params
#define TPB 256     // threads per block = 8 wave32

typedef uint32_t u32x4 __attribute__((ext_vector_type(4)));
typedef uint32_t u32x8 __attribute__((ext_vector_type(8)));

// ---------------------------------------------------------------------------
// Kernel 1: radial symmetry layer.
// One block handles 256 consecutive atoms of one batch. The batch's whole
// coordinate table X[b] (2048*3 f32 = 24KB) is staged into LDS with a single
// Tensor Data Mover descriptor (CDNA5 TDM), then each thread owns one atom
// and loops over its 64 neighbors with LDS gathers.
// Writes layer[b][n][c] (c = type*12 + l) to `out`.
// ---------------------------------------------------------------------------
__global__ __launch_bounds__(TPB) void rsf_layer_kernel(
    const float* __restrict__ X,
    const int*   __restrict__ Nbrs,
    const int*   __restrict__ NbrsZ,
    const float* __restrict__ rc,
    const float* __restrict__ rs,
    const float* __restrict__ re,
    float*       __restrict__ out)
{
    __shared__ float sX[NN * 3];   // 24 KB coordinate stage

    const int blocksPerBatch = NN / TPB;            // 8
    const int b  = (int)blockIdx.x / blocksPerBatch;
    const int n0 = ((int)blockIdx.x % blocksPerBatch) * TPB;
    const int t  = (int)threadIdx.x;

    // ---- TDM: one descriptor copies X[b,:,:] (6144 f32) global -> LDS ----
    if (t < 32) {  // wave 0 issues the DMA (TDM ignores EXEC; one wave only)
        const uint32_t ldsOff = (uint32_t)(uintptr_t)(void*)&sX[0];
        const uint64_t ga     = (uint64_t)(uintptr_t)(const void*)
                                (X + (size_t)b * (NN * 3));
        u32x4 g0;
        g0[0] = 1u;                                   // count = 1 valid D#
        g0[1] = ldsOff;                               // lds_addr
        g0[2] = (uint32_t)ga;                         // global_addr[31:0]
        g0[3] = ((uint32_t)(ga >> 32) & 0x01FFFFFFu)  // global_addr[56:32]
              | (2u << 30);                           // type = 2 ("image")
        const uint32_t d0 = NN * 3;                   // 6144 elements
        u32x8 g1;
        g1[0] = (2u << 16);               // workgroup_mask=0, data_size=4B
        g1[1] = (d0 & 0xFFFFu) << 16;     // tensor_dim0[15:0]
        g1[2] = (d0 >> 16) | (1u << 16);  // tensor_dim0[31:16] | tensor_dim1=1
        g1[3] = (d0 & 0xFFFFu) << 16;     // tensor_dim1 hi=0 | tile_dim0=6144
        g1[4] = 1u;                       // tile_dim1=1, tile_dim2=0
        g1[5] = d0;                       // tensor_dim0_stride[31:0]
        g1[6] = 0u;                       // stride0 hi | stride1 lo
        g1[7] = 0u;                       // stride1 hi
        asm volatile("tensor_load_to_lds %0, %1" :: "s"(g0), "s"(g1) : "memory");
#if __has_builtin(__builtin_amdgcn_s_wait_tensorcnt)
        __builtin_amdgcn_s_wait_tensorcnt(0);
#else
        asm volatile("s_wait_tensorcnt 0x0" ::: "memory");
#endif
    }
    __syncthreads();

    // ---- uniform per-l parameters (scalar loads, kept in registers) ----
    float rcl[LL], rsl[LL], rel[LL], prl[LL];
    float rcmax = 0.0f;
#pragma unroll
    for (int l = 0; l < LL; ++l) {
        rcl[l] = rc[l];
        rsl[l] = rs[l];
        rel[l] = re[l];
        prl[l] = 3.14159265358979323846f / rcl[l];
        rcmax  = fmaxf(rcmax, rcl[l]);
    }

    const int n = n0 + t;
    const float x0 = sX[3 * n + 0];
    const float y0 = sX[3 * n + 1];
    const float z0 = sX[3 * n + 2];

    float acc[CH];
#pragma unroll
    for (int c = 0; c < CH; ++c) acc[c] = 0.0f;

    const int* __restrict__ nb = Nbrs  + ((size_t)b * NN + n) * MM;
    const int* __restrict__ zb = NbrsZ + ((size_t)b * NN + n) * MM;

    for (int mg = 0; mg < MM; mg += 4) {
        const int4 ki = *(const int4*)(nb + mg);
        const int4 zi = *(const int4*)(zb + mg);
        const int kk[4] = {ki.x, ki.y, ki.z, ki.w};
        const int zz[4] = {zi.x, zi.y, zi.z, zi.w};
#pragma unroll
        for (int j = 0; j < 4; ++j) {
            const int k = kk[j];
            const float dx = sX[3 * k + 0] - x0;
            const float dy = sX[3 * k + 1] - y0;
            const float dz = sX[3 * k + 2] - z0;
            const float R  = sqrtf(dx * dx + dy * dy + dz * dz);
            if (R <= rcmax) {  // skipped only when the whole wave is far
                const int z = zz[j];
                const float w0 = (z == 1) ? 1.0f : 0.0f;  // H
                const float w1 = (z == 6) ? 1.0f : 0.0f;  // C
                const float w2 = (z == 7) ? 1.0f : 0.0f;  // N
                const float w3 = (z == 8) ? 1.0f : 0.0f;  // O
#pragma unroll
                for (int l = 0; l < LL; ++l) {
                    const float d  = R - rsl[l];
                    const float K  = __expf(-rel[l] * d * d);
                    const float fc = (R <= rcl[l])
                                   ? 0.5f * (__cosf(prl[l] * R) + 1.0f)
                                   : 0.0f;
                    const float v = K * fc;
                    acc[0 * LL + l] += w0 * v;
                    acc[1 * LL + l] += w1 * v;
                    acc[2 * LL + l] += w2 * v;
                    acc[3 * LL + l] += w3 * v;
                }
            }
        }
    }

    float* __restrict__ o = out + ((size_t)b * NN + n) * CH;  // 192B aligned
#pragma unroll
    for (int c = 0; c < CH; c += 4) {
        float4 v;
        v.x = acc[c + 0]; v.y = acc[c + 1]; v.z = acc[c + 2]; v.w = acc[c + 3];
        *(float4*)(o + c) = v;
    }
}

// ---------------------------------------------------------------------------
// Kernel 2: batch-norm over the batch axis (biased variance), in place.
// One thread per (n, c); consecutive threads -> consecutive addresses for
// each b, so all 16 reads and writes are fully coalesced.
// ---------------------------------------------------------------------------
__global__ __launch_bounds__(TPB) void bn_kernel(float* __restrict__ layer)
{
    const int idx = (int)blockIdx.x * TPB + (int)threadIdx.x;  // (n*CH + c)
    if (idx >= NN * CH) return;

    float v[BB];
    float s = 0.0f;
#pragma unroll
    for (int b = 0; b < BB; ++b) {
        v[b] = layer[(size_t)b * (NN * CH) + idx];
        s += v[b];
    }
    const float mean = s * (1.0f / BB);
    float var = 0.0f;
#pragma unroll
    for (int b = 0; b < BB; ++b) {
        const float d = v[b] - mean;
        var += d * d;
    }
    var *= (1.0f / BB);
    const float inv = rsqrtf(var + 1e-3f);
#pragma unroll
    for (int b = 0; b < BB; ++b) {
        layer[(size_t)b * (NN * CH) + idx] = (v[b] - mean) * inv;
    }
}

// ---------------------------------------------------------------------------
extern "C" void kernel_launch(void* const* d_in, const int* in_sizes, int n_in,
                              void* d_out, int out_size, void* d_ws, size_t ws_size,
                              hipStream_t stream)
{
    (void)in_sizes; (void)n_in; (void)out_size; (void)d_ws; (void)ws_size;

    const float* X    = (const float*)d_in[0];
    const int*   Nbrs = (const int*)  d_in[1];
    const int*   Z    = (const int*)  d_in[2];
    const float* rc   = (const float*)d_in[3];
    const float* rs   = (const float*)d_in[4];
    const float* re   = (const float*)d_in[5];
    float* out = (float*)d_out;

    rsf_layer_kernel<<<BB * (NN / TPB), TPB, 0, stream>>>(X, Nbrs, Z, rc, rs, re, out);
    bn_kernel<<<(NN * CH + TPB - 1) / TPB, TPB, 0, stream>>>(out);
}